// BDHAttention_10187662426758
// MI455X (gfx1250) — compile-verified
//
#include <hip/hip_runtime.h>
#include <hip/hip_bf16.h>
#include <math.h>

// ---------------------------------------------------------------------------
// BDH linear-attention forward for MI455X (gfx1250, wave32, WMMA).
//   B=4, T=4096, C=1024, H=16, D=64.  fp32 via V_WMMA_F32_16X16X4_F32
//   (memory-bound at 23.3 TB/s -> no precision drop needed).
// Pipeline:
//   1) gemm_kt:  qkv = x @ w_qkv^T   (16384x3072x1024), LDS double-buffered,
//                staged with GLOBAL_LOAD_ASYNC_TO_LDS_B128 (ASYNCcnt).
//   2) rope:     RoPE+ReLU on q,k; split to (B,H,T,D) layout.
//   3) scan:     chunked (L=64) recurrence; Q/K/V chunks double-buffered via
//                the Tensor Data Mover (tensor_load_to_lds + s_wait_tensorcnt 1
//                -> DMA of chunk c+1 overlaps compute of chunk c).
//   4) gemm_kt:  out = att @ w_proj^T (16384x1024x1024).
// ---------------------------------------------------------------------------

typedef __attribute__((ext_vector_type(2))) float v2f;
typedef __attribute__((ext_vector_type(8))) float v8f;
typedef __attribute__((ext_vector_type(4))) unsigned int u32x4;
typedef __attribute__((ext_vector_type(8))) unsigned int u32x8;

__device__ __forceinline__ v8f wmma4(v2f a, v2f b, v8f c) {
  return __builtin_amdgcn_wmma_f32_16x16x4_f32(
      false, a, false, b, (short)0, c, false, false);
}

// Raw LDS byte offset of a __shared__ object (generic -> AS3 -> int).
__device__ __forceinline__ unsigned int lds_off(const void* p) {
  return (unsigned int)(uintptr_t)(__attribute__((address_space(3))) const void*)p;
}

// Async 16B global -> LDS copy (per-lane addresses), tracked by ASYNCcnt.
__device__ __forceinline__ void async_ld_b128(unsigned int ldsAddr, const float* g) {
  asm volatile("global_load_async_to_lds_b128 %0, %1, off"
               :: "v"(ldsAddr), "v"(g) : "memory");
}
__device__ __forceinline__ void wait_async0() {
  asm volatile("s_wait_asynccnt 0x0" ::: "memory");
}

// TDM: DMA a contiguous run of `ndw` dwords (ndw <= 65535) from global to LDS.
// D# group0/group1 built per CDNA5 ISA; 1-row 2D tile, 4B elems.
__device__ __forceinline__ void tdm_load_1d(unsigned int ldsAddr,
                                            const float* g, unsigned int ndw) {
  unsigned long long ga = (unsigned long long)(uintptr_t)g;
  u32x4 g0;
  g0[0] = 1u;                                        // count=1, user descriptor
  g0[1] = ldsAddr;                                   // lds_addr (bytes)
  g0[2] = (unsigned int)(ga & 0xFFFFFFFFull);        // global_addr[31:0]
  g0[3] = ((unsigned int)(ga >> 32) & 0x1FFFFFFu)    // global_addr[56:32]
          | (2u << 30);                              // type=2 ("image")
  u32x8 g1;
  g1[0] = (2u << 16);                 // wg_mask=0, data_size=2 (4 bytes)
  g1[1] = (ndw & 0xFFFFu) << 16;      // atomic_barrier_addr=0 | tensor_dim0.lo16
  g1[2] = ((ndw >> 16) & 0xFFFFu)     // tensor_dim0.hi16
          | (1u << 16);               // tensor_dim1 = 1 (lo16)
  g1[3] = (ndw & 0xFFFFu) << 16;      // tensor_dim1.hi16=0 | tile_dim0 = ndw
  g1[4] = 1u;                         // tile_dim1 = 1, tile_dim2 = 0
  g1[5] = ndw;                        // tensor_dim0_stride lo32
  g1[6] = 0u;                         // stride hi16 | tensor_dim1_stride lo16
  g1[7] = 0u;                         // tensor_dim1_stride hi32
  asm volatile("tensor_load_to_lds %0, %1" :: "s"(g0), "s"(g1) : "memory");
}

// ---------------------------------------------------------------------------
// Tiled GEMM: C[m,n] = sum_k A[m,k] * B[n,k]  (both K-contiguous).
// Block: 256 threads = 8 waves; tile BM=128 x BN=64, K-step 32, double-
// buffered LDS staged via async b128 copies. Wave w -> rows [16w,16w+16).
// Requires: M%128==0, N%64==0, K%32==0, rows 16B-aligned (K%4==0).
// ---------------------------------------------------------------------------
#define APAD 36  // padded row length (floats): bank-conflict-free fragments

__global__ __launch_bounds__(256)
void gemm_kt_kernel(const float* __restrict__ A, const float* __restrict__ Bm,
                    float* __restrict__ Cm, int M, int N, int K) {
  __shared__ float Alds[2 * 128 * APAD];
  __shared__ float Blds[2 * 64 * APAD];

  const int tid    = threadIdx.x;
  const int lane   = tid & 31;
  const int wave   = tid >> 5;
  const int ntiles = N >> 6;
  const int mt     = blockIdx.x / ntiles;
  const int nt     = blockIdx.x - mt * ntiles;
  const int m0     = mt << 7;              // 128-row tile
  const int n0     = nt << 6;              // 64-col tile
  const int lrow   = lane & 15;
  const int kh     = (lane >> 4) << 1;     // K sub-offset 0 / 2

  const unsigned int aOff = lds_off(Alds);
  const unsigned int bOff = lds_off(Blds);

  // Stage one 32-deep K slab into buffer `buf` (async, no wait here).
  auto stage = [&](int buf, int k0) {
    const unsigned int aB = aOff + (unsigned)buf * (128 * APAD * 4);
#pragma unroll
    for (int i = 0; i < 4; ++i) {          // A: 128 rows x 8 chunks = 1024
      const int c = tid + (i << 8);
      const int row = c >> 3, kc = c & 7;
      async_ld_b128(aB + row * (APAD * 4) + kc * 16,
                    A + (size_t)(m0 + row) * K + k0 + kc * 4);
    }
    const unsigned int bB = bOff + (unsigned)buf * (64 * APAD * 4);
#pragma unroll
    for (int i = 0; i < 2; ++i) {          // B: 64 rows x 8 chunks = 512
      const int c = tid + (i << 8);
      const int row = c >> 3, kc = c & 7;
      async_ld_b128(bB + row * (APAD * 4) + kc * 16,
                    Bm + (size_t)(n0 + row) * K + k0 + kc * 4);
    }
  };

  v8f acc[4] = {};
  const int nK = K >> 5;
  int buf = 0;
  stage(0, 0);
  wait_async0();
  __syncthreads();

  for (int kt = 0; kt < nK; ++kt) {
    if (kt + 1 < nK) stage(buf ^ 1, (kt + 1) << 5);   // prefetch next slab

    const float* Ar = Alds + buf * (128 * APAD) + (wave * 16 + lrow) * APAD;
    const float* Br = Blds + buf * (64 * APAD) + lrow * APAD;
#pragma unroll
    for (int k = 0; k < 32; k += 4) {
      v2f a = *(const v2f*)(Ar + k + kh);
#pragma unroll
      for (int j = 0; j < 4; ++j) {
        v2f b = *(const v2f*)(Br + j * 16 * APAD + k + kh);
        acc[j] = wmma4(a, b, acc[j]);
      }
    }
    wait_async0();       // prefetched slab landed
    __syncthreads();     // everyone done with `buf`; safe to flip
    buf ^= 1;
  }

  // C/D layout: VGPR r -> row r (lanes 0-15) / row 8+r (lanes 16-31), N=lane&15
  const int rbase = (lane >> 4) << 3;
#pragma unroll
  for (int j = 0; j < 4; ++j) {
    const int col = n0 + j * 16 + lrow;
#pragma unroll
    for (int r = 0; r < 8; ++r) {
      Cm[(size_t)(m0 + wave * 16 + rbase + r) * N + col] = acc[j][r];
    }
  }
}

// ---------------------------------------------------------------------------
// RoPE + ReLU on q,k; pass-through v; repack (B,T,H,D) -> (B,H,T,D).
// ---------------------------------------------------------------------------
__global__ __launch_bounds__(256)
void rope_relu_split_kernel(const float* __restrict__ qkv,
                            float* __restrict__ qs, float* __restrict__ ks,
                            float* __restrict__ vs) {
  const int i = blockIdx.x * blockDim.x + threadIdx.x;   // over 2^24 elements
  const int d = i & 63;
  const int h = (i >> 6) & 15;
  const int t = (i >> 10) & 4095;
  const int b = i >> 22;

  const size_t row = (size_t)((b << 12) + t) * 3072;
  const int   col  = (h << 6) + d;
  const int   dp   = (d < 32) ? d + 32 : d - 32;
  const int   colp = (h << 6) + dp;

  const float qv = qkv[row + col];
  const float qp = qkv[row + colp];
  const float kv = qkv[row + 1024 + col];
  const float kp = qkv[row + 1024 + colp];
  const float vv = qkv[row + 2048 + col];

  const float rotq = (d < 32) ? -qp : qp;
  const float rotk = (d < 32) ? -kp : kp;

  const float invf = __powf(10000.f, -(float)((d & 31) << 1) * (1.0f / 64.0f));
  const float ang  = (float)t * invf;
  const float cs = __cosf(ang);
  const float sn = __sinf(ang);

  const size_t o = (size_t)(((b << 4) + h) * 4096 + t) * 64 + d;
  qs[o] = fmaxf(qv * cs + rotq * sn, 0.f);
  ks[o] = fmaxf(kv * cs + rotk * sn, 0.f);
  vs[o] = vv;
}

// ---------------------------------------------------------------------------
// Chunked linear-attention scan. One block (8 waves) per (b,h) sequence.
//   P = eta*stricttril(Q K^T);  O = Q @ S_in + P @ V;  S += eta * K^T V
// Wave (wr,wc) = (wave>>1, wave&1) owns rows [16wr,16wr+16) x cols
// [32wc,32wc+32) of every 64x64 product (disjoint sigma ownership).
// Q/K/V chunks DMA'd by the TDM with double buffering: chunk c+1 is issued
// before computing chunk c; s_wait_tensorcnt(1) retires only the older DMA.
// Static LDS: 16KB (S) + 2 x 48KB (Q,K,V) = 112KB (<= 320KB/WGP).
// ---------------------------------------------------------------------------
__global__ __launch_bounds__(256)
void scan_kernel(const float* __restrict__ qs, const float* __restrict__ ks,
                 const float* __restrict__ vs, const float* __restrict__ sigma0,
                 const float* __restrict__ etaPtr,
                 float* __restrict__ att, float* __restrict__ sigmaOut) {
  __shared__ float Slds[64 * 64];
  __shared__ float Qc[2 * 64 * 64];   // active half also parks masked P
  __shared__ float Kc[2 * 64 * 64];
  __shared__ float Vc[2 * 64 * 64];

  const float eta = etaPtr[0];
  const int bh    = blockIdx.x;          // 0..63
  const int tid   = threadIdx.x;         // 0..255
  const int wave  = tid >> 5;            // 0..7
  const int lane  = tid & 31;
  const int lrow  = lane & 15;
  const int kh    = (lane >> 4) << 1;
  const int m0    = (wave >> 1) << 4;    // row stripe
  const int nc0   = (wave & 1) << 5;     // column half (0 or 32)
  const int rbase = (lane >> 4) << 3;

  const unsigned int qOff = lds_off(Qc);
  const unsigned int kOff = lds_off(Kc);
  const unsigned int vOff = lds_off(Vc);

  const size_t seq = (size_t)bh * (4096 * 64);

  // prologue: DMA chunk 0 into buffer 0, stage sigma0 meanwhile
  if (wave == 0)      tdm_load_1d(qOff, qs + seq, 4096);
  else if (wave == 1) tdm_load_1d(kOff, ks + seq, 4096);
  else if (wave == 2) tdm_load_1d(vOff, vs + seq, 4096);
  for (int i = tid; i < 4096; i += 256)
    Slds[i] = sigma0[(size_t)bh * 4096 + i];

  const int b = bh >> 4, h = bh & 15;
  float* attBase = att + (size_t)b * 4096 * 1024 + (h << 6);
  __syncthreads();   // sigma staged (TDM retired per-chunk below)

  int buf = 0;
  for (int c = 0; c < 64; ++c) {
    if (c + 1 < 64) {                  // prefetch next chunk into other buffer
      const size_t offn = seq + (size_t)(c + 1) * 4096;
      const unsigned int bo = (unsigned)(buf ^ 1) * 16384;  // bytes
      if (wave == 0)      tdm_load_1d(qOff + bo, qs + offn, 4096);
      else if (wave == 1) tdm_load_1d(kOff + bo, ks + offn, 4096);
      else if (wave == 2) tdm_load_1d(vOff + bo, vs + offn, 4096);
      __builtin_amdgcn_s_wait_tensorcnt(1);   // chunk c landed; c+1 in flight
    } else {
      __builtin_amdgcn_s_wait_tensorcnt(0);
    }
    __syncthreads();

    float* Qb = Qc + buf * 4096;
    float* Kb = Kc + buf * 4096;
    float* Vb = Vc + buf * 4096;

    // Phase A: O = Q @ S_in ; P = Q K^T   (16x32 piece per wave)
    v8f o[2] = {};
    v8f p[2] = {};
#pragma unroll 1
    for (int k = 0; k < 64; k += 4) {
      v2f a = *(const v2f*)&Qb[(m0 + lrow) * 64 + k + kh];
#pragma unroll
      for (int j = 0; j < 2; ++j) {
        const int cb = nc0 + j * 16 + lrow;
        v2f bS;                                        // B = S (K,N layout)
        bS.x = Slds[(k + kh) * 64 + cb];
        bS.y = Slds[(k + kh + 1) * 64 + cb];
        o[j] = wmma4(a, bS, o[j]);
        v2f bK = *(const v2f*)&Kb[cb * 64 + k + kh];   // B = K^T
        p[j] = wmma4(a, bK, p[j]);
      }
    }
    __syncthreads();   // all waves done reading Qb and Slds

    // causal mask (strict lower) + eta scale; park P in the active Q buffer
#pragma unroll
    for (int j = 0; j < 2; ++j) {
      const int col = nc0 + j * 16 + lrow;
#pragma unroll
      for (int r = 0; r < 8; ++r) {
        const int row = m0 + rbase + r;
        Qb[row * 64 + col] = (col < row) ? p[j][r] * eta : 0.f;
      }
    }
    __syncthreads();

    // Phase B: O += P @ V
#pragma unroll 1
    for (int k = 0; k < 64; k += 4) {
      v2f a = *(const v2f*)&Qb[(m0 + lrow) * 64 + k + kh];
#pragma unroll
      for (int j = 0; j < 2; ++j) {
        const int cb = nc0 + j * 16 + lrow;
        v2f bV;
        bV.x = Vb[(k + kh) * 64 + cb];
        bV.y = Vb[(k + kh + 1) * 64 + cb];
        o[j] = wmma4(a, bV, o[j]);
      }
    }
    // write O chunk: att[b, c*64+row, h*64+col]
#pragma unroll
    for (int j = 0; j < 2; ++j) {
      const int col = nc0 + j * 16 + lrow;
#pragma unroll
      for (int r = 0; r < 8; ++r) {
        const int row = m0 + rbase + r;
        attBase[(size_t)((c << 6) + row) * 1024 + col] = o[j][r];
      }
    }

    // Phase C: U = K^T V ; S[stripe] += eta * U
    v8f u[2] = {};
#pragma unroll 1
    for (int k = 0; k < 64; k += 4) {
      v2f a;                                           // A = K^T (d rows)
      a.x = Kb[(k + kh) * 64 + m0 + lrow];
      a.y = Kb[(k + kh + 1) * 64 + m0 + lrow];
#pragma unroll
      for (int j = 0; j < 2; ++j) {
        const int cb = nc0 + j * 16 + lrow;
        v2f bV;
        bV.x = Vb[(k + kh) * 64 + cb];
        bV.y = Vb[(k + kh + 1) * 64 + cb];
        u[j] = wmma4(a, bV, u[j]);
      }
    }
#pragma unroll
    for (int j = 0; j < 2; ++j) {
      const int col = nc0 + j * 16 + lrow;
#pragma unroll
      for (int r = 0; r < 8; ++r) {
        const int row = m0 + rbase + r;   // this wave's owned sigma piece
        Slds[row * 64 + col] += eta * u[j][r];
      }
    }
    __syncthreads();   // buf free for the prefetch two iterations ahead
    buf ^= 1;
  }

  for (int i = tid; i < 4096; i += 256)
    sigmaOut[(size_t)bh * 4096 + i] = Slds[i];
}

// ---------------------------------------------------------------------------
extern "C" void kernel_launch(void* const* d_in, const int* in_sizes, int n_in,
                              void* d_out, int out_size, void* d_ws, size_t ws_size,
                              hipStream_t stream) {
  const float* x      = (const float*)d_in[0];
  const float* sigma0 = (const float*)d_in[1];
  const float* w_qkv  = (const float*)d_in[2];
  const float* w_proj = (const float*)d_in[3];
  const float* eta    = (const float*)d_in[4];

  float* ws   = (float*)d_ws;
  float* qkv  = ws;                         // 50331648 floats (192 MB)
  float* qs   = ws + 50331648;              // 16777216 floats
  float* ks   = qs + 16777216;
  float* vs   = ks + 16777216;
  float* att  = ws;                         // reuse qkv region after rope

  float* out      = (float*)d_out;          // 16777216 floats
  float* sigmaOut = out + 16777216;         //   262144 floats

  // 1) qkv = x @ w_qkv^T : tiles (16384/128) x (3072/64) = 6144 blocks
  gemm_kt_kernel<<<6144, 256, 0, stream>>>(x, w_qkv, qkv, 16384, 3072, 1024);

  // 2) RoPE + ReLU + layout split (2^24 elements)
  rope_relu_split_kernel<<<65536, 256, 0, stream>>>(qkv, qs, ks, vs);

  // 3) chunked recurrent scan: one block per (b,h)
  scan_kernel<<<64, 256, 0, stream>>>(qs, ks, vs, sigma0, eta, att, sigmaOut);

  // 4) out = att @ w_proj^T : (16384/128) x (1024/64) = 2048 blocks
  gemm_kt_kernel<<<2048, 256, 0, stream>>>(att, w_proj, out, 16384, 1024, 1024);
}